// FeatureProjection_11338713661700
// MI455X (gfx1250) — compile-verified
//
#include <hip/hip_runtime.h>

// out[i] = in[i] * scale[0] over 44,040,192 fp32 elements (8x21x512x512).
// Pure streaming op (0.125 FLOP/byte): HBM roofline ~15us at 23.3 TB/s.
// Strategy: block-tiled b128 nontemporal loads/stores with compile-time
// strides so addressing folds into immediate offsets (saddr + 32-bit voffset).

typedef float __attribute__((ext_vector_type(4))) f4;

#define THREADS 256
#define UNROLL  4
#define TILE    (THREADS * UNROLL)   // 1024 float4s = 16 KB per block

__global__ __launch_bounds__(THREADS) void feature_scale_kernel(
    const float* __restrict__ in, const float* __restrict__ scale_p,
    float* __restrict__ out, int n4, long long n) {
  // Wave-uniform -> s_load_b32 on the SMEM path.
  const float s = scale_p[0];

  const unsigned tid       = threadIdx.x;
  const unsigned blockBase = blockIdx.x * (unsigned)TILE;  // float4 units
  const unsigned i0        = blockBase + tid;

  const char* __restrict__ inb  = (const char*)in;
  char* __restrict__       outb = (char*)out;

  if (blockBase + TILE <= (unsigned)n4) {
    // Full tile: uniform (scalar) branch, no EXEC masking.
    // One address; +THREADS strides are 4096-byte compile-time constants ->
    // immediate offsets on global_load_b128 / global_store_b128.
    const f4* __restrict__ p = (const f4*)(inb + (size_t)(i0 * 16u));
    f4* __restrict__       q = (f4*)(outb + (size_t)(i0 * 16u));
    f4 v0 = __builtin_nontemporal_load(p);
    f4 v1 = __builtin_nontemporal_load(p + THREADS);
    f4 v2 = __builtin_nontemporal_load(p + 2 * THREADS);
    f4 v3 = __builtin_nontemporal_load(p + 3 * THREADS);
    v0 *= s; v1 *= s; v2 *= s; v3 *= s;
    __builtin_nontemporal_store(v0, q);
    __builtin_nontemporal_store(v1, q + THREADS);
    __builtin_nontemporal_store(v2, q + 2 * THREADS);
    __builtin_nontemporal_store(v3, q + 3 * THREADS);
  } else {
    // Partial tile (last block only, if n4 % TILE != 0).
#pragma unroll
    for (int k = 0; k < UNROLL; ++k) {
      unsigned i = i0 + (unsigned)k * THREADS;
      if (i < (unsigned)n4) {
        f4 v = __builtin_nontemporal_load((const f4*)(inb + (size_t)(i * 16u)));
        v *= s;
        __builtin_nontemporal_store(v, (f4*)(outb + (size_t)(i * 16u)));
      }
    }
  }

  // Scalar tail for n % 4 != 0 (none for this shape; last block handles it).
  if (blockIdx.x == gridDim.x - 1) {
    for (long long j = (long long)n4 * 4 + tid; j < n; j += THREADS) {
      out[j] = in[j] * s;
    }
  }
}

extern "C" void kernel_launch(void* const* d_in, const int* in_sizes, int n_in,
                              void* d_out, int out_size, void* d_ws, size_t ws_size,
                              hipStream_t stream) {
  (void)in_sizes; (void)n_in; (void)d_ws; (void)ws_size;
  const float* feature = (const float*)d_in[0];
  const float* scale   = (const float*)d_in[1];
  float* out           = (float*)d_out;

  const long long n  = (long long)out_size;   // 44,040,192
  const int       n4 = (int)(n / 4);          // 11,010,048 float4s

  int blocks = (n4 + TILE - 1) / TILE;        // 10,752 blocks (exact tiles)
  if (blocks < 1) blocks = 1;

  feature_scale_kernel<<<blocks, THREADS, 0, stream>>>(feature, scale, out, n4, n);
}